// SuperpointAttention_v2_2319282340202
// MI455X (gfx1250) — compile-verified
//
#include <hip/hip_runtime.h>
#include <hip/hip_bf16.h>
#include <math.h>

// Problem constants (match reference)
constexpr int NB = 4;           // batches
constexpr int NP = 8192;        // points per batch
constexpr int NK = 8;           // neighbors
constexpr int NC = 256;         // channels
constexpr int NROWS = NB * NP;  // 32768
constexpr float INV_SCALE = 0.35355339059327373f; // 1/sqrt(8)
constexpr float LN_EPS = 1e-5f;

typedef __attribute__((ext_vector_type(2))) float v2f;
typedef __attribute__((ext_vector_type(8))) float v8f;
typedef unsigned int u32x4 __attribute__((ext_vector_type(4)));
typedef int          i32x4 __attribute__((ext_vector_type(4)));
typedef int          i32x8 __attribute__((ext_vector_type(8)));

#if __has_builtin(__builtin_amdgcn_tensor_load_to_lds) && __has_builtin(__builtin_amdgcn_s_wait_tensorcnt)
#define USE_TDM 1
#else
#define USE_TDM 0
#endif

#if USE_TDM
// ---------------------------------------------------------------------------
// Tensor Data Mover: DMA a 2D tile (data_size = 4B) from global into LDS.
// D# layout per CDNA5 ISA §8.3-8.5.  Optional LDS padding: after every
// 2^(pad_interval+1) dwords stored, skip (pad_amount+1) dwords in LDS.
// Issued by one wave; completion tracked with TENSORcnt.
// This toolchain exposes the 6-arg builtin:
//   (uint32x4 g0, int32x8 g1, int32x4 g2, int32x4 g3, int32x8 extra, int cpol)
// ---------------------------------------------------------------------------
__device__ __forceinline__ void tdm_load_to_lds(unsigned lds_addr, const void* gptr,
                                                unsigned tensor_d0, unsigned tensor_d1,
                                                unsigned tile_d0, unsigned tile_d1,
                                                unsigned d0_stride,
                                                unsigned pad_interval, unsigned pad_amount,
                                                bool pad_en) {
    const unsigned long long ga = (unsigned long long)(uintptr_t)gptr;
    u32x4 g0;
    g0[0] = 1u;                                               // count=1, user descriptor
    g0[1] = lds_addr;                                         // LDS byte address
    g0[2] = (unsigned)ga;                                     // global_addr[31:0]
    g0[3] = (unsigned)((ga >> 32) & 0x01ffffffu) | (2u << 30);// global_addr[56:32] | type=2
    unsigned dw0 = (2u << 16);                                // data_size = 4 bytes
    if (pad_en) dw0 |= (1u << 20) | (pad_interval << 22) | (pad_amount << 25);
    i32x8 g1;
    g1[0] = (int)dw0;                                         // wg_mask=0 (not in cluster)
    g1[1] = (int)((tensor_d0 & 0xffffu) << 16);               // tensor_dim0[15:0] @ [63:48]
    g1[2] = (int)((tensor_d0 >> 16) | ((tensor_d1 & 0xffffu) << 16));
    g1[3] = (int)((tensor_d1 >> 16) | (tile_d0 << 16));       // tile_dim0 @ [127:112]
    g1[4] = (int)(tile_d1 & 0xffffu);                         // tile_dim1, tile_dim2=0
    g1[5] = (int)d0_stride;                                   // tensor_dim0_stride[31:0]
    g1[6] = 0;
    g1[7] = 0;
    const i32x4 z4 = {0, 0, 0, 0};                            // groups 2/3 unused (2D)
    const i32x8 z8 = {0, 0, 0, 0, 0, 0, 0, 0};
    __builtin_amdgcn_tensor_load_to_lds(g0, g1, z4, z4, z8, 0);
}
#endif

// ---------------------------------------------------------------------------
// Kernel 1: fp32 WMMA GEMM  Out(M x 256) = A(M x 256) * W(256 x 256)
// One wave -> one 16x16 tile, 64 chained V_WMMA_F32_16X16X4_F32 over K.
// A strip staged by TDM directly into the padded LDS layout (hardware pad).
// grid = (M/16, 2), block = 256 (8 waves).
// ---------------------------------------------------------------------------
__global__ void gemm_f32_wmma(const float* __restrict__ A,
                              const float* __restrict__ W,
                              float* __restrict__ Out) {
    __shared__ float As[16][NC + 4];   // +4 dwords/row: conflict-free A-frag reads
    const int row0 = blockIdx.x * 16;
    const int wave = threadIdx.x >> 5;          // wave32
    const int lane = threadIdx.x & 31;
    const int col0 = blockIdx.y * 128 + wave * 16;

#if USE_TDM
    if (threadIdx.x == 0) {
        // 16 x 256 fp32 tile; pad 4 dwords after each 256-dword row -> As stride 260
        tdm_load_to_lds((unsigned)(uintptr_t)&As[0][0], A + (size_t)row0 * NC,
                        /*tensor_d0=*/NC, /*tensor_d1=*/(unsigned)(NROWS - row0),
                        /*tile_d0=*/NC, /*tile_d1=*/16, /*d0_stride=*/NC,
                        /*pad_interval=*/7, /*pad_amount=*/3, /*pad_en=*/true);
        __builtin_amdgcn_s_wait_tensorcnt(0);
    }
    __syncthreads();
#else
    for (int t = threadIdx.x; t < 16 * NC; t += 256) {
        const int r = t >> 8;
        const int k = t & (NC - 1);
        As[r][k] = A[(size_t)(row0 + r) * NC + k];
    }
    __syncthreads();
#endif

    const int m    = lane & 15;      // row within tile (A) / col within tile (B,C)
    const int half = lane >> 4;      // lane half selects K pair / M half

    v8f acc = {};                    // C = 0
    #pragma unroll 4
    for (int k0 = 0; k0 < NC; k0 += 4) {
        // A frag: lanes 0-15 hold K={k0,k0+1}, lanes 16-31 hold K={k0+2,k0+3}
        v2f a, b;
        a.x = As[m][k0 + 2 * half];
        a.y = As[m][k0 + 2 * half + 1];
        // B frag: row striped across lanes; VGPR0 rows {k0, k0+2}, VGPR1 rows {k0+1, k0+3}
        b.x = W[(size_t)(k0 + 2 * half)     * NC + col0 + m];
        b.y = W[(size_t)(k0 + 2 * half + 1) * NC + col0 + m];
        // D = A*B + C   (8 args: neg_a, A, neg_b, B, c_mod, C, reuse_a, reuse_b)
        acc = __builtin_amdgcn_wmma_f32_16x16x4_f32(false, a, false, b,
                                                    (short)0, acc, false, false);
    }

    // C/D layout: VGPR r -> M = r + 8*half, N = lane&15
    #pragma unroll
    for (int r = 0; r < 8; ++r) {
        const int mr = r + half * 8;
        Out[(size_t)(row0 + mr) * NC + col0 + m] = acc[r];
    }
}

// ---------------------------------------------------------------------------
// Kernel 2: P(M x 256) = xyz(M x 3) @ W_coord(3 x 256)   (K=3: plain VALU FMA)
// ---------------------------------------------------------------------------
__global__ void coord_proj(const float* __restrict__ coords,   // (M,4) [bid,x,y,z]
                           const float* __restrict__ Wc,       // (3,256)
                           float* __restrict__ P) {
    const int i = blockIdx.x;
    const int c = threadIdx.x;
    const float x = coords[(size_t)i * 4 + 1];
    const float y = coords[(size_t)i * 4 + 2];
    const float z = coords[(size_t)i * 4 + 3];
    P[(size_t)i * NC + c] = fmaf(x, Wc[c], fmaf(y, Wc[NC + c], z * Wc[2 * NC + c]));
}

// ---------------------------------------------------------------------------
// Kernel 3: brute-force kNN (K=8) per batch.
// Double-buffered 2048-point (32KB) candidate tiles DMA'd into LDS by the TDM
// (wave 0 issues tile t+1 while all 8 waves scan tile t from LDS). Candidate
// reads are uniform-address ds_load_b128 broadcasts. Sorted top-8 lives
// entirely in registers (unrolled bubble insert; strict '<' matches top_k's
// lower-index tie-break).
// grid = (NP/256, NB), block = 256.
// ---------------------------------------------------------------------------
constexpr int KTILE = 2048;                       // candidates per tile

__global__ void knn_topk(const float* __restrict__ coords,  // (NB*NP, 4)
                         int* __restrict__ idx_out) {       // (NB*NP, 8) global row ids
    const int b = blockIdx.y;
    const int i = blockIdx.x * 256 + threadIdx.x;           // point within batch
    const float* cb = coords + (size_t)b * NP * 4;

    __shared__ float4 tile[2][KTILE];                       // 2 x 32KB

    const float qx = cb[(size_t)i * 4 + 1];
    const float qy = cb[(size_t)i * 4 + 2];
    const float qz = cb[(size_t)i * 4 + 3];

    float bd[NK];
    int   bi[NK];
    #pragma unroll
    for (int k = 0; k < NK; ++k) { bd[k] = 3.4e38f; bi[k] = 0; }

    constexpr int NT = NP / KTILE;                          // 4 tiles

#if USE_TDM
    if (threadIdx.x == 0) {                                 // prologue: tile 0
        tdm_load_to_lds((unsigned)(uintptr_t)&tile[0][0], cb,
                        /*tensor_d0=*/KTILE * 4, /*tensor_d1=*/1,
                        /*tile_d0=*/KTILE * 4, /*tile_d1=*/1, /*d0_stride=*/KTILE * 4,
                        0, 0, false);
    }
#endif

    for (int t = 0; t < NT; ++t) {
#if USE_TDM
        if (threadIdx.x == 0)
            __builtin_amdgcn_s_wait_tensorcnt(0);           // tile t landed in LDS
        __syncthreads();
        if (threadIdx.x == 0 && t + 1 < NT) {               // overlap: DMA tile t+1
            tdm_load_to_lds((unsigned)(uintptr_t)&tile[(t + 1) & 1][0],
                            cb + (size_t)(t + 1) * KTILE * 4,
                            KTILE * 4, 1, KTILE * 4, 1, KTILE * 4, 0, 0, false);
        }
#else
        __syncthreads();
        for (int u = threadIdx.x; u < KTILE; u += 256) {
            const float* src = cb + (size_t)(t * KTILE + u) * 4;
            tile[t & 1][u] = make_float4(src[0], src[1], src[2], src[3]);
        }
        __syncthreads();
#endif
        const float4* cur = tile[t & 1];
        const int jbase = t * KTILE;

        #pragma unroll 4
        for (int u = 0; u < KTILE; ++u) {
            const float4 p = cur[u];                        // broadcast ds_load_b128
            const float dx = qx - p.y;
            const float dy = qy - p.z;
            const float dz = qz - p.w;
            const float d2 = fmaf(dx, dx, fmaf(dy, dy, dz * dz));
            if (d2 < bd[NK - 1]) {                          // rarely taken
                float dcur = d2;
                int   icur = jbase + u;
                #pragma unroll
                for (int pp = 0; pp < NK; ++pp) {           // bubble into sorted list
                    if (dcur < bd[pp]) {
                        const float td = bd[pp]; bd[pp] = dcur; dcur = td;
                        const int   ti = bi[pp]; bi[pp] = icur; icur = ti;
                    }
                }
            }
        }
        __syncthreads();                                    // done reading before reuse
    }

    const size_t base = ((size_t)b * NP + i) * NK;
    #pragma unroll
    for (int k = 0; k < NK; ++k)
        idx_out[base + k] = b * NP + bi[k];                 // global row index
}

// ---------------------------------------------------------------------------
// Kernel 4: gather + gated softmax over K + weighted sum + residual + LayerNorm.
// One block per point, one thread per channel (softmax over K is thread-local).
// grid = NROWS, block = 256.
// ---------------------------------------------------------------------------
__global__ void attn_combine_ln(const float* __restrict__ F1,   // X@W_feat
                                const float* __restrict__ F2,   // X@W_ft
                                const float* __restrict__ P,    // xyz@W_coord
                                const float* __restrict__ feats,
                                const int* __restrict__ idx,
                                const float* __restrict__ b_coord,
                                const float* __restrict__ b_feat,
                                const float* __restrict__ b_ft,
                                const float* __restrict__ gamma,
                                const float* __restrict__ beta,
                                float* __restrict__ out) {
    const int i = blockIdx.x;
    const int c = threadIdx.x;

    __shared__ int   nidx[NK];
    __shared__ float ssum[NC];
    __shared__ float ssq[NC];

    if (threadIdx.x < NK) nidx[threadIdx.x] = idx[(size_t)i * NK + threadIdx.x];
    __syncthreads();

    const float pc  = P [(size_t)i * NC + c];
    const float f1c = F1[(size_t)i * NC + c];
    const float bc = b_coord[c], bf = b_feat[c], bt = b_ft[c];

    // prefetch neighbor rows (global_prefetch_b8 on gfx1250)
    #pragma unroll
    for (int k = 0; k < NK; ++k) {
        const size_t n = (size_t)nidx[k] * NC + c;
        __builtin_prefetch(&F1[n], 0, 3);
        __builtin_prefetch(&F2[n], 0, 3);
    }

    float lg[NK], vv[NK];
    #pragma unroll
    for (int k = 0; k < NK; ++k) {
        const size_t n = (size_t)nidx[k] * NC + c;
        const float sc = P[n]  - pc  + bc;        // rc @ W_coord + b_coord
        const float sf = F1[n] - f1c + bf;        // rf @ W_feat  + b_feat
        lg[k] = sc * sf * INV_SCALE;
        vv[k] = F2[n] + bt;                       // nf @ W_ft + b_ft
    }

    float mx = lg[0];
    #pragma unroll
    for (int k = 1; k < NK; ++k) mx = fmaxf(mx, lg[k]);
    float s = 0.0f, acc = 0.0f;
    #pragma unroll
    for (int k = 0; k < NK; ++k) {
        const float e = __expf(lg[k] - mx);
        s   += e;
        acc  = fmaf(e, vv[k], acc);
    }
    const float o = acc / s + feats[(size_t)i * NC + c];  // residual

    // LayerNorm over 256 channels (LDS tree reduction)
    ssum[c] = o;
    ssq[c]  = o * o;
    __syncthreads();
    #pragma unroll
    for (int off = 128; off > 0; off >>= 1) {
        if (c < off) {
            ssum[c] += ssum[c + off];
            ssq[c]  += ssq[c + off];
        }
        __syncthreads();
    }
    const float mu  = ssum[0] * (1.0f / NC);
    const float var = ssq[0] * (1.0f / NC) - mu * mu;
    const float inv = rsqrtf(var + LN_EPS);
    out[(size_t)i * NC + c] = (o - mu) * inv * gamma[c] + beta[c];
}

// ---------------------------------------------------------------------------
// Host-side launcher
// ---------------------------------------------------------------------------
extern "C" void kernel_launch(void* const* d_in, const int* in_sizes, int n_in,
                              void* d_out, int out_size, void* d_ws, size_t ws_size,
                              hipStream_t stream) {
    (void)in_sizes; (void)n_in; (void)out_size; (void)ws_size;

    const float* features = (const float*)d_in[0];
    const float* coords   = (const float*)d_in[1];
    const float* W_coord  = (const float*)d_in[2];
    const float* b_coord  = (const float*)d_in[3];
    const float* W_feat   = (const float*)d_in[4];
    const float* b_feat   = (const float*)d_in[5];
    const float* W_ft     = (const float*)d_in[6];
    const float* b_ft     = (const float*)d_in[7];
    const float* gamma    = (const float*)d_in[8];
    const float* beta     = (const float*)d_in[9];
    float* out = (float*)d_out;

    // Workspace partition (total ~101.7 MB)
    char* ws = (char*)d_ws;
    const size_t matBytes = (size_t)NROWS * NC * sizeof(float);  // 32 MB each
    float* F1 = (float*)ws;  ws += matBytes;
    float* F2 = (float*)ws;  ws += matBytes;
    float* P  = (float*)ws;  ws += matBytes;
    int* nbr  = (int*)ws;    // NROWS*NK ints = 1 MB

    // 1) fp32 WMMA GEMMs: F1 = X@W_feat, F2 = X@W_ft
    dim3 gGemm(NROWS / 16, NC / 128);
    gemm_f32_wmma<<<gGemm, 256, 0, stream>>>(features, W_feat, F1);
    gemm_f32_wmma<<<gGemm, 256, 0, stream>>>(features, W_ft, F2);

    // 2) coord projection P = xyz @ W_coord
    coord_proj<<<NROWS, NC, 0, stream>>>(coords, W_coord, P);

    // 3) brute-force kNN per batch (TDM double-buffered candidate tiles)
    knn_topk<<<dim3(NP / 256, NB), 256, 0, stream>>>(coords, nbr);

    // 4) gather + softmax + residual + LayerNorm
    attn_combine_ln<<<NROWS, NC, 0, stream>>>(F1, F2, P, features, nbr,
                                              b_coord, b_feat, b_ft,
                                              gamma, beta, out);
}